// MessagePass_88210038325909
// MI455X (gfx1250) — compile-verified
//
#include <hip/hip_runtime.h>
#include <hip/hip_bf16.h>

typedef float v2f __attribute__((ext_vector_type(2)));
typedef float v8f __attribute__((ext_vector_type(8)));

#define B_  16
#define C_  128
#define H_  64
#define W_  256
#define KK_ 1152                 // C_ * 9
#define CSTRIDE (H_ * W_)        // 16384
#define BSTRIDE (C_ * H_ * W_)   // 2097152

#define KC  72                   // K chunk: 8 input channels * 9 taps
#define MT  64                   // block tile M
#define NT  64                   // block tile N (never crosses a batch row: 256 % 64 == 0)
#define A_PAD 73                 // LDS row pitch (odd -> conflict-free strided reads)
#define B_PAD 65

// dst[b,o,w] = relu( (sum_{i,k} Wt[o,i*9+k]*src[b,i,w+k-4]) * scale[o] + shift[o] ) + addv[b,o,w]
// src/addv/dst are base pointers already offset by h*W_ (slice selection).
__global__ __launch_bounds__(256)
void MessagePass_step_wmma(const float* __restrict__ src,
                           const float* __restrict__ addv,
                           float* __restrict__ dst,
                           const float* __restrict__ Wt,   // [128][1152]
                           const float* __restrict__ ss)   // [0..127]=scale, [128..255]=shift
{
    __shared__ float As[MT * A_PAD];   // weights  [m][k]
    __shared__ float Bs[KC * B_PAD];   // patches  [k][n]

    const int tid  = threadIdx.x;
    const int lane = tid & 31;
    const int hi   = lane >> 4;        // 0: lanes 0-15, 1: lanes 16-31
    const int l16  = lane & 15;
    const int wv   = tid >> 5;         // wave id 0..7

    const int m0 = blockIdx.y * MT;            // 0 or 64
    const int n0 = blockIdx.x * NT;            // 0..4032
    const int b  = n0 / W_;                    // batch (constant per block)
    const int w0 = n0 % W_;

    const int mt  = wv & 3;                    // wave's M-tile (16 rows)
    const int ntp = (wv >> 2) * 2;             // wave's first N-tile (of 2)

    v8f acc0 = {};
    v8f acc1 = {};

    for (int kc0 = 0; kc0 < KK_; kc0 += KC) {
        __syncthreads();
        // ---- stage A: 64 x 72 weight block (A[m][kk] == Wt[o][i*9+k], contiguous) ----
        for (int idx = tid; idx < MT * KC; idx += 256) {
            int mr = idx / KC, kk = idx % KC;
            As[mr * A_PAD + kk] = Wt[(m0 + mr) * KK_ + kc0 + kk];
        }
        // ---- stage B: 72 x 64 im2col patch block with zero padding ----
        for (int idx = tid; idx < KC * NT; idx += 256) {
            int kk  = idx / NT, col = idx % NT;
            int kg  = kc0 + kk;
            int ic  = kg / 9;
            int tap = kg % 9;
            int w   = w0 + col + tap - 4;
            float v = 0.0f;
            if (w >= 0 && w < W_)
                v = src[(size_t)b * BSTRIDE + (size_t)ic * CSTRIDE + w];
            Bs[kk * B_PAD + col] = v;
        }
        __syncthreads();

        const float* Arow = &As[(mt * 16 + l16) * A_PAD];
        #pragma unroll
        for (int kk = 0; kk < KC; kk += 4) {
            const int kb = kk + 2 * hi;        // lanes 0-15: K=kk,kk+1 ; lanes 16-31: K=kk+2,kk+3
            v2f a;
            a.x = Arow[kb];
            a.y = Arow[kb + 1];
            v2f bf0, bf1;                      // B frag: row per VGPR, N across lanes
            bf0.x = Bs[kb * B_PAD + ntp * 16 + l16];
            bf0.y = Bs[(kb + 1) * B_PAD + ntp * 16 + l16];
            bf1.x = Bs[kb * B_PAD + (ntp + 1) * 16 + l16];
            bf1.y = Bs[(kb + 1) * B_PAD + (ntp + 1) * 16 + l16];
            acc0 = __builtin_amdgcn_wmma_f32_16x16x4_f32(false, a, false, bf0,
                                                         (short)0, acc0, false, false);
            acc1 = __builtin_amdgcn_wmma_f32_16x16x4_f32(false, a, false, bf1,
                                                         (short)0, acc1, false, false);
        }
    }

    // ---- epilogue: BN fold + ReLU + residual add ----
    const float* scale = ss;
    const float* shift = ss + C_;
    #pragma unroll
    for (int j = 0; j < 8; ++j) {
        const int M = mt * 16 + j + 8 * hi;    // C/D layout: vgpr j -> M=j (lo) / j+8 (hi)
        const int o = m0 + M;
        const float sc = scale[o];
        const float sh = shift[o];
        {
            const int w = w0 + ntp * 16 + l16;
            const size_t off = (size_t)b * BSTRIDE + (size_t)o * CSTRIDE + w;
            float y = acc0[j] * sc + sh;
            y = y > 0.0f ? y : 0.0f;
            dst[off] = y + addv[off];
        }
        {
            const int w = w0 + (ntp + 1) * 16 + l16;
            const size_t off = (size_t)b * BSTRIDE + (size_t)o * CSTRIDE + w;
            float y = acc1[j] * sc + sh;
            y = y > 0.0f ? y : 0.0f;
            dst[off] = y + addv[off];
        }
    }
}

// out slice h=0 := x slice h=0  ((b,c) flattens cleanly since b-stride = C_*H_*W_)
__global__ void MessagePass_copy_slice(const float* __restrict__ x, float* __restrict__ out)
{
    int e = blockIdx.x * blockDim.x + threadIdx.x;
    if (e >= B_ * C_ * W_) return;
    int w  = e % W_;
    int bc = e / W_;
    size_t off = (size_t)bc * CSTRIDE + w;
    out[off] = x[off];
}

// fold BN (eval mode) into per-channel scale/shift
__global__ void MessagePass_bn_prep(const float* __restrict__ g, const float* __restrict__ be,
                                    const float* __restrict__ rm, const float* __restrict__ rv,
                                    float* __restrict__ ss)
{
    int c = blockIdx.x * blockDim.x + threadIdx.x;
    if (c < C_) {
        float sc = g[c] * rsqrtf(rv[c] + 1e-5f);
        ss[c]      = sc;
        ss[C_ + c] = be[c] - rm[c] * sc;
    }
}

extern "C" void kernel_launch(void* const* d_in, const int* in_sizes, int n_in,
                              void* d_out, int out_size, void* d_ws, size_t ws_size,
                              hipStream_t stream)
{
    const float* x     = (const float*)d_in[0];   // [16,128,64,256]
    const float* cw    = (const float*)d_in[1];   // [128,128,9]
    const float* gamma = (const float*)d_in[2];
    const float* beta  = (const float*)d_in[3];
    const float* rmean = (const float*)d_in[4];
    const float* rvar  = (const float*)d_in[5];
    float* out = (float*)d_out;                   // [16,128,64,256]
    float* ss  = (float*)d_ws;                    // 256 floats scale|shift

    (void)in_sizes; (void)n_in; (void)out_size; (void)ws_size;

    MessagePass_bn_prep<<<1, 128, 0, stream>>>(gamma, beta, rmean, rvar, ss);
    MessagePass_copy_slice<<<(B_ * C_ * W_ + 255) / 256, 256, 0, stream>>>(x, out);

    dim3 grid((B_ * W_) / NT, C_ / MT);           // (64, 2) = 128 workgroups per step

    // forward scan: new[h] = f(new[h-1]) + x[h]   (new[] lives in out)
    for (int h = 1; h < H_; ++h)
        MessagePass_step_wmma<<<grid, 256, 0, stream>>>(
            out + (size_t)(h - 1) * W_, x + (size_t)h * W_, out + (size_t)h * W_, cw, ss);

    // backward scan: out[h] = f(out[h+1]) + new[h]   (in-place residual read)
    for (int h = H_ - 2; h >= 0; --h)
        MessagePass_step_wmma<<<grid, 256, 0, stream>>>(
            out + (size_t)(h + 1) * W_, out + (size_t)h * W_, out + (size_t)h * W_, cw, ss);
}